// GridFeaturesAndText_33895881900477
// MI455X (gfx1250) — compile-verified
//
#include <hip/hip_runtime.h>
#include <math.h>

typedef __attribute__((ext_vector_type(2))) float v2f;
typedef __attribute__((ext_vector_type(8))) float v8f;

constexpr int kQ = 768;   // QUEST_DIM
constexpr int kC = 2048;  // channels
constexpr int kS = 361;   // 19*19 spatial
constexpr int kB = 64;    // batch

// ---------------------------------------------------------------------------
// Kernel 1: q[b,c] = text[b,:] . W[c,:] + bias[c]   via V_WMMA_F32_16X16X4_F32
// One wave computes a 16(M)x16(N) tile of q, looping K=768 in steps of 4.
// A-frag (16x4 f32): lane(l%16)=M row, half-wave(l/16) selects K pair -> float2
// B-frag (4x16 f32): lane(l%16)=N col (W row, since B = W^T), same K mapping
// C/D  (16x16 f32): VGPR v, lane l -> (M = m0 + v + 8*(l/16), N = n0 + l%16)
// ---------------------------------------------------------------------------
__global__ __launch_bounds__(256) void qgemm_wmma(const float* __restrict__ text,
                                                  const float* __restrict__ W,
                                                  const float* __restrict__ bias,
                                                  float* __restrict__ q) {
    const int wid  = threadIdx.x >> 5;
    const int lane = threadIdx.x & 31;
    const int half = lane >> 4;
    const int lm   = lane & 15;

    const int tile = blockIdx.x * 8 + wid;   // 512 tiles total: 4 (M) x 128 (N)
    const int m0 = (tile >> 7) << 4;
    const int n0 = (tile & 127) << 4;

    const float* arow = text + (m0 + lm) * kQ + half * 2;
    const float* brow = W    + (n0 + lm) * kQ + half * 2;

    v8f acc = {};
#pragma unroll 4
    for (int k = 0; k < kQ; k += 4) {
        v2f a  = *(const v2f*)(arow + k);
        v2f bf = *(const v2f*)(brow + k);
        acc = __builtin_amdgcn_wmma_f32_16x16x4_f32(
            /*neg_a=*/false, a, /*neg_b=*/false, bf,
            /*c_mod=*/(short)0, acc, /*reuse_a=*/false, /*reuse_b=*/false);
    }

    const float bv = bias[n0 + lm];
#pragma unroll
    for (int v = 0; v < 8; ++v) {
        q[(m0 + v + 8 * half) * kC + (n0 + lm)] = acc[v] + bv;
    }
}

// ---------------------------------------------------------------------------
// Kernel 2: sims[b,s] = sum_c q[b,c]*img[b,c,s]; then softmax over s -> wgt.
// One block per batch row. q row staged in LDS; each thread owns one s, so
// img loads are coalesced across the 361 spatial positions for every c.
// ---------------------------------------------------------------------------
__global__ __launch_bounds__(384) void sims_softmax(const float* __restrict__ img,
                                                    const float* __restrict__ q,
                                                    float* __restrict__ wgt) {
    __shared__ float qlds[kC];
    __shared__ float red[512];

    const int b   = blockIdx.x;
    const int tid = threadIdx.x;

    for (int i = tid; i < kC; i += 384) qlds[i] = q[b * kC + i];
    __syncthreads();

    float acc = 0.f;
    if (tid < kS) {
        const float* p = img + (size_t)b * (kC * kS) + tid;
#pragma unroll 4
        for (int c = 0; c < kC; ++c) acc += qlds[c] * p[c * kS];
    }

    // --- max over s ---
    red[tid] = (tid < kS) ? acc : -INFINITY;
    if (tid < 128) red[384 + tid] = -INFINITY;
    __syncthreads();
    for (int off = 256; off >= 1; off >>= 1) {
        if (tid < off) { float o = red[tid + off]; if (o > red[tid]) red[tid] = o; }
        __syncthreads();
    }
    const float mx = red[0];
    __syncthreads();

    // --- sum of exp ---
    const float e = (tid < kS) ? __expf(acc - mx) : 0.f;
    red[tid] = e;
    if (tid < 128) red[384 + tid] = 0.f;
    __syncthreads();
    for (int off = 256; off >= 1; off >>= 1) {
        if (tid < off) red[tid] += red[tid + off];
        __syncthreads();
    }

    const float inv = 1.f / red[0];
    if (tid < kS) wgt[b * kS + tid] = e * inv;
}

// ---------------------------------------------------------------------------
// Kernel 3: out[b,c] = sum_s img[b,c,s]*wgt[b,s].
// One wave per channel row; 32 lanes stride over s (coalesced img reads),
// weights broadcast from LDS, wave-level shuffle reduction.
// Grid: (kC/64, kB); each of 8 waves handles 8 channel rows.
// ---------------------------------------------------------------------------
__global__ __launch_bounds__(256) void pool(const float* __restrict__ img,
                                            const float* __restrict__ wgt,
                                            float* __restrict__ out) {
    __shared__ float wl[kS];

    const int b    = blockIdx.y;
    const int tid  = threadIdx.x;
    const int wid  = tid >> 5;
    const int lane = tid & 31;

    for (int i = tid; i < kS; i += 256) wl[i] = wgt[b * kS + i];
    __syncthreads();

    const int cbase = blockIdx.x * 64 + wid * 8;
#pragma unroll
    for (int i = 0; i < 8; ++i) {
        const int c = cbase + i;
        const float* row = img + (size_t)b * (kC * kS) + c * kS;
        float acc = 0.f;
        for (int sb = 0; sb < kS; sb += 32) {
            const int s = sb + lane;
            if (s < kS) acc += row[s] * wl[s];
        }
#pragma unroll
        for (int off = 16; off > 0; off >>= 1) acc += __shfl_down(acc, off, 32);
        if (lane == 0) out[b * kC + c] = acc;
    }
}

// ---------------------------------------------------------------------------
extern "C" void kernel_launch(void* const* d_in, const int* in_sizes, int n_in,
                              void* d_out, int out_size, void* d_ws, size_t ws_size,
                              hipStream_t stream) {
    const float* text = (const float*)d_in[0];   // [64, 768]
    const float* img  = (const float*)d_in[1];   // [64, 2048, 19, 19]
    const float* W    = (const float*)d_in[2];   // [2048, 768]
    const float* bias = (const float*)d_in[3];   // [2048]
    float* out = (float*)d_out;                  // [64, 2048]

    float* q   = (float*)d_ws;                   // kB*kC floats (512 KB)
    float* wgt = q + kB * kC;                    // kB*kS floats (~92 KB)

    qgemm_wmma  <<<64,              256, 0, stream>>>(text, W, bias, q);
    sims_softmax<<<kB,              384, 0, stream>>>(img, q, wgt);
    pool        <<<dim3(kC/64, kB), 256, 0, stream>>>(img, wgt, out);
}